// GroupPopEncoder_69776038690918
// MI455X (gfx1250) — compile-verified
//
#include <hip/hip_runtime.h>

typedef float v2f __attribute__((ext_vector_type(2)));
typedef float v8f __attribute__((ext_vector_type(8)));

#define DIMV 8
#define NHEADS 4
#define ATT_SCALE 0.35355339059327373f  // 1/sqrt(8): reference divides by sqrt(dim_V)

// ---------------------------------------------------------------------------
// WMMA: D = A(16x4,f32) * B(4x16,f32) + C(16x16,f32)
// ---------------------------------------------------------------------------
__device__ __forceinline__ v8f wmma4(v2f a, v2f b, v8f c) {
  return __builtin_amdgcn_wmma_f32_16x16x4_f32(false, a, false, b, (short)0, c,
                                               false, false);
}

// ---------------------------------------------------------------------------
// Stage one (8,din) weight matrix + bias into LDS, zero-padded to 16x8 / 16.
// wl[n*8+k] = (n<8 && k<din) ? w[n*din+k] : 0 ;  bl[n] = (n<8) ? b[n] : 0
// ---------------------------------------------------------------------------
__device__ __forceinline__ void stage_w(float* wl, float* bl, const float* w,
                                        const float* b, int din, int t, int nt) {
  for (int i = t; i < 128; i += nt) {
    const int n = i >> 3, k = i & 7;
    wl[i] = (n < 8 && k < din) ? w[n * din + k] : 0.0f;
  }
  for (int i = t; i < 16; i += nt) bl[i] = (i < 8) ? b[i] : 0.0f;
}

// ---------------------------------------------------------------------------
// dst[r][0..7] = src[r][:] @ w^T + bias  for rows [tile0*16, tile1*16).
// wl/bl are LDS zero-padded weights. nChunk=1 (din<=4, src stride 4, padded)
// or 2 (din=8, stride 8). Whole wave active (WMMA EXEC requirement).
// A (16x4): M=lane&15, K pair {0,1}/{2,3} by lane half. B (4x16): N=lane&15,
// same K split. C/D (16x16): N=lane&15, M=vgpr+8*(lane>=16).
// ---------------------------------------------------------------------------
__device__ __forceinline__ void proj_tiles_wmma(const float* src, int srcStride,
                                                int nChunk, const float* wl,
                                                const float* bl, float* dst,
                                                int lane, int tile0, int tile1) {
  const int n = lane & 15, hi = lane >> 4;
  const float bv = bl[n];
  v2f bf0 = {wl[n * 8 + hi * 2], wl[n * 8 + hi * 2 + 1]};
  v2f bf1 = {0.0f, 0.0f};
  if (nChunk > 1) bf1 = (v2f){wl[n * 8 + 4 + hi * 2], wl[n * 8 + 4 + hi * 2 + 1]};
  for (int tile = tile0; tile < tile1; ++tile) {
    v8f acc;
#pragma unroll
    for (int r = 0; r < 8; ++r) acc[r] = bv;
    const int row = tile * 16 + n;
    v2f a = {src[row * srcStride + hi * 2], src[row * srcStride + hi * 2 + 1]};
    acc = wmma4(a, bf0, acc);
    if (nChunk > 1) {
      v2f a2 = {src[row * srcStride + 4 + hi * 2],
                src[row * srcStride + 4 + hi * 2 + 1]};
      acc = wmma4(a2, bf1, acc);
    }
    if (n < DIMV) {
#pragma unroll
      for (int r = 0; r < 8; ++r)
        dst[(tile * 16 + r + hi * 8) * DIMV + n] = acc[r];
    }
  }
}

// H = H + relu(H @ wo^T + bo), in place, rows [tile0*16, tile1*16), stride 8.
__device__ __forceinline__ void ff_tiles_wmma(float* H, const float* wl,
                                              const float* bl, int lane,
                                              int tile0, int tile1) {
  const int n = lane & 15, hi = lane >> 4;
  const float bv = bl[n];
  v2f bf0 = {wl[n * 8 + hi * 2], wl[n * 8 + hi * 2 + 1]};
  v2f bf1 = {wl[n * 8 + 4 + hi * 2], wl[n * 8 + 4 + hi * 2 + 1]};
  for (int tile = tile0; tile < tile1; ++tile) {
    v8f acc;
#pragma unroll
    for (int r = 0; r < 8; ++r) acc[r] = bv;
    const int row = tile * 16 + n;
    v2f a0 = {H[row * 8 + hi * 2], H[row * 8 + hi * 2 + 1]};
    acc = wmma4(a0, bf0, acc);
    v2f a1 = {H[row * 8 + 4 + hi * 2], H[row * 8 + 4 + hi * 2 + 1]};
    acc = wmma4(a1, bf1, acc);
    if (n < DIMV) {
#pragma unroll
      for (int r = 0; r < 8; ++r) {
        const int idx = (tile * 16 + r + hi * 8) * DIMV + n;
        const float lin = acc[r];
        H[idx] = H[idx] + (lin > 0.0f ? lin : 0.0f);
      }
    }
  }
}

// thread t owns query row t: Q[t] += softmax(Q[t].Kh^T * scale) @ V per head.
// K/V rows loaded as float4 pairs (ds_load_b128); all 4 heads per pass.
__device__ __forceinline__ void attn_row_inplace(float* Q, const float* K,
                                                 const float* V, int N, int t) {
  float4 qa = *(const float4*)(Q + t * 8);
  float4 qb = *(const float4*)(Q + t * 8 + 4);
  float m0 = -1e30f, m1 = -1e30f, m2 = -1e30f, m3 = -1e30f;
  for (int j = 0; j < N; ++j) {
    float4 ka = *(const float4*)(K + j * 8);
    float4 kb = *(const float4*)(K + j * 8 + 4);
    m0 = fmaxf(m0, qa.x * ka.x + qa.y * ka.y);
    m1 = fmaxf(m1, qa.z * ka.z + qa.w * ka.w);
    m2 = fmaxf(m2, qb.x * kb.x + qb.y * kb.y);
    m3 = fmaxf(m3, qb.z * kb.z + qb.w * kb.w);
  }
  m0 *= ATT_SCALE; m1 *= ATT_SCALE; m2 *= ATT_SCALE; m3 *= ATT_SCALE;
  float s0 = 0, s1 = 0, s2 = 0, s3 = 0;
  float a0 = 0, a1 = 0, a2 = 0, a3 = 0, a4 = 0, a5 = 0, a6 = 0, a7 = 0;
  for (int j = 0; j < N; ++j) {
    float4 ka = *(const float4*)(K + j * 8);
    float4 kb = *(const float4*)(K + j * 8 + 4);
    float4 va = *(const float4*)(V + j * 8);
    float4 vb = *(const float4*)(V + j * 8 + 4);
    float e0 = __expf((qa.x * ka.x + qa.y * ka.y) * ATT_SCALE - m0);
    float e1 = __expf((qa.z * ka.z + qa.w * ka.w) * ATT_SCALE - m1);
    float e2 = __expf((qb.x * kb.x + qb.y * kb.y) * ATT_SCALE - m2);
    float e3 = __expf((qb.z * kb.z + qb.w * kb.w) * ATT_SCALE - m3);
    s0 += e0; s1 += e1; s2 += e2; s3 += e3;
    a0 += e0 * va.x; a1 += e0 * va.y;
    a2 += e1 * va.z; a3 += e1 * va.w;
    a4 += e2 * vb.x; a5 += e2 * vb.y;
    a6 += e3 * vb.z; a7 += e3 * vb.w;
  }
  float4 oa = {qa.x + a0 / s0, qa.y + a1 / s0, qa.z + a2 / s1, qa.w + a3 / s1};
  float4 ob = {qb.x + a4 / s2, qb.y + a5 / s2, qb.z + a6 / s3, qb.w + a7 / s3};
  *(float4*)(Q + t * 8) = oa;
  *(float4*)(Q + t * 8 + 4) = ob;
}

struct MabW { const float *qw, *qb, *kw, *kb, *vw, *vb, *ow, *ob; };
struct ArgsA {
  const float* x; MabW e0; MabW e1; const float* S; MabW dm;
  const float* lw; const float* lb; float* out; float* zmid;
};
struct ArgsB {
  const float* zmid; MabW e0; MabW e1; const float* S; MabW dm;
  const float* lw; const float* lb; float* out;
};

// ---------------------------------------------------------------------------
// Phase A: one block per set (512 blocks, 256 threads = 8 waves).
// enc1[0] -> enc1[1] -> PMA(dec1) -> lin ; writes z1 slice of out and zmid.
// ---------------------------------------------------------------------------
__global__ __launch_bounds__(256) void phaseA(ArgsA p) {
  __shared__ float Xs[256 * 4];      // x padded to 4 cols (cols 2,3 = 0)
  __shared__ float T0[256 * 8];
  __shared__ float T1[256 * 8];
  __shared__ float Kb[256 * 8];
  __shared__ float Vb[256 * 8];
  __shared__ float sc[NHEADS * 256];
  __shared__ float WL[10 * 128];     // zero-padded weights
  __shared__ float BL[10 * 16];      // zero-padded biases
  __shared__ float small[24];        // [0:8) sS  [8:16) H  [16:24) O

  const int t = threadIdx.x, lane = t & 31, wave = t >> 5;
  const int s = blockIdx.x;  // set index 0..511

  // async DMA x tile (2 floats/lane) into padded LDS rows (ASYNCcnt tracked)
  {
    unsigned ldsoff = (unsigned)(size_t)(Xs + t * 4);
    const float* g = p.x + (size_t)s * 512 + t * 2;
    asm volatile("global_load_async_to_lds_b64 %0, %1, off"
                 :: "v"(ldsoff), "v"(g) : "memory");
  }
  Xs[t * 4 + 2] = 0.0f;
  Xs[t * 4 + 3] = 0.0f;

  // stage all projection weights (zero-padded) while the DMA is in flight
  stage_w(WL + 0 * 128, BL + 0 * 16, p.e0.qw, p.e0.qb, 2, t, 256);
  stage_w(WL + 1 * 128, BL + 1 * 16, p.e0.kw, p.e0.kb, 2, t, 256);
  stage_w(WL + 2 * 128, BL + 2 * 16, p.e0.vw, p.e0.vb, 2, t, 256);
  stage_w(WL + 3 * 128, BL + 3 * 16, p.e0.ow, p.e0.ob, 8, t, 256);
  stage_w(WL + 4 * 128, BL + 4 * 16, p.e1.qw, p.e1.qb, 8, t, 256);
  stage_w(WL + 5 * 128, BL + 5 * 16, p.e1.kw, p.e1.kb, 8, t, 256);
  stage_w(WL + 6 * 128, BL + 6 * 16, p.e1.vw, p.e1.vb, 8, t, 256);
  stage_w(WL + 7 * 128, BL + 7 * 16, p.e1.ow, p.e1.ob, 8, t, 256);
  stage_w(WL + 8 * 128, BL + 8 * 16, p.dm.kw, p.dm.kb, 8, t, 256);
  stage_w(WL + 9 * 128, BL + 9 * 16, p.dm.vw, p.dm.vb, 8, t, 256);

  asm volatile("s_wait_asynccnt 0x0" ::: "memory");
  __syncthreads();

  // ---- enc1[0] (din=2, 1 K-chunk) ----
  proj_tiles_wmma(Xs, 4, 1, WL + 0 * 128, BL + 0 * 16, T0, lane, wave * 2, wave * 2 + 2);
  proj_tiles_wmma(Xs, 4, 1, WL + 1 * 128, BL + 1 * 16, Kb, lane, wave * 2, wave * 2 + 2);
  proj_tiles_wmma(Xs, 4, 1, WL + 2 * 128, BL + 2 * 16, Vb, lane, wave * 2, wave * 2 + 2);
  __syncthreads();
  attn_row_inplace(T0, Kb, Vb, 256, t);
  __syncthreads();
  ff_tiles_wmma(T0, WL + 3 * 128, BL + 3 * 16, lane, wave * 2, wave * 2 + 2);
  __syncthreads();

  // ---- enc1[1] (din=8, 2 K-chunks) ----
  proj_tiles_wmma(T0, 8, 2, WL + 4 * 128, BL + 4 * 16, T1, lane, wave * 2, wave * 2 + 2);
  proj_tiles_wmma(T0, 8, 2, WL + 5 * 128, BL + 5 * 16, Kb, lane, wave * 2, wave * 2 + 2);
  proj_tiles_wmma(T0, 8, 2, WL + 6 * 128, BL + 6 * 16, Vb, lane, wave * 2, wave * 2 + 2);
  __syncthreads();
  attn_row_inplace(T1, Kb, Vb, 256, t);
  __syncthreads();
  ff_tiles_wmma(T1, WL + 7 * 128, BL + 7 * 16, lane, wave * 2, wave * 2 + 2);
  __syncthreads();

  // ---- dec1: PMA (Nq = 1) ----
  proj_tiles_wmma(T1, 8, 2, WL + 8 * 128, BL + 8 * 16, Kb, lane, wave * 2, wave * 2 + 2);
  proj_tiles_wmma(T1, 8, 2, WL + 9 * 128, BL + 9 * 16, Vb, lane, wave * 2, wave * 2 + 2);
  if (t < 8) {  // q projection of seed S (1x8)
    float acc = p.dm.qb[t];
    for (int k = 0; k < 8; ++k) acc += p.S[k] * p.dm.qw[t * 8 + k];
    small[t] = acc;
  }
  __syncthreads();
#pragma unroll
  for (int h = 0; h < NHEADS; ++h)
    sc[h * 256 + t] =
        (small[2 * h] * Kb[t * 8 + 2 * h] + small[2 * h + 1] * Kb[t * 8 + 2 * h + 1]) * ATT_SCALE;
  __syncthreads();
  if (t < NHEADS) {
    const int h = t;
    float m = -1e30f;
    for (int j = 0; j < 256; ++j) m = fmaxf(m, sc[h * 256 + j]);
    float ssum = 0.0f, a0 = 0.0f, a1 = 0.0f;
    for (int j = 0; j < 256; ++j) {
      float e = __expf(sc[h * 256 + j] - m);
      ssum += e;
      a0 += e * Vb[j * 8 + 2 * h];
      a1 += e * Vb[j * 8 + 2 * h + 1];
    }
    const float inv = 1.0f / ssum;
    small[8 + 2 * h]     = small[2 * h]     + a0 * inv;
    small[8 + 2 * h + 1] = small[2 * h + 1] + a1 * inv;
  }
  __syncthreads();
  if (t < 8) {  // O = H + relu(H @ wo^T + bo)
    float acc = p.dm.ob[t];
    for (int k = 0; k < 8; ++k) acc += small[8 + k] * p.dm.ow[t * 8 + k];
    small[16 + t] = small[8 + t] + fmaxf(acc, 0.0f);
  }
  __syncthreads();
  if (t < 8) {  // z = lin(O)
    float acc = p.lb[t];
    for (int k = 0; k < 8; ++k) acc += small[16 + k] * p.lw[t * 8 + k];
    p.zmid[s * 8 + t] = acc;
    const int b = s >> 5, g = s & 31;
    p.out[b * 264 + g * 8 + t] = acc;  // z1 slice
  }
}

// ---------------------------------------------------------------------------
// Phase B: one block per batch (16 blocks, 32 threads = 1 wave), 32 tokens.
// ---------------------------------------------------------------------------
__global__ __launch_bounds__(32) void phaseB(ArgsB p) {
  __shared__ float T0[32 * 8];
  __shared__ float T1[32 * 8];
  __shared__ float Kb[32 * 8];
  __shared__ float Vb[32 * 8];
  __shared__ float sc[NHEADS * 32];
  __shared__ float WL[10 * 128];
  __shared__ float BL[10 * 16];
  __shared__ float small[24];

  const int t = threadIdx.x;  // 0..31, single wave
  const int b = blockIdx.x;
#pragma unroll
  for (int j = 0; j < 8; ++j) T0[t * 8 + j] = p.zmid[(b * 32 + t) * 8 + j];

  stage_w(WL + 0 * 128, BL + 0 * 16, p.e0.qw, p.e0.qb, 8, t, 32);
  stage_w(WL + 1 * 128, BL + 1 * 16, p.e0.kw, p.e0.kb, 8, t, 32);
  stage_w(WL + 2 * 128, BL + 2 * 16, p.e0.vw, p.e0.vb, 8, t, 32);
  stage_w(WL + 3 * 128, BL + 3 * 16, p.e0.ow, p.e0.ob, 8, t, 32);
  stage_w(WL + 4 * 128, BL + 4 * 16, p.e1.qw, p.e1.qb, 8, t, 32);
  stage_w(WL + 5 * 128, BL + 5 * 16, p.e1.kw, p.e1.kb, 8, t, 32);
  stage_w(WL + 6 * 128, BL + 6 * 16, p.e1.vw, p.e1.vb, 8, t, 32);
  stage_w(WL + 7 * 128, BL + 7 * 16, p.e1.ow, p.e1.ob, 8, t, 32);
  stage_w(WL + 8 * 128, BL + 8 * 16, p.dm.kw, p.dm.kb, 8, t, 32);
  stage_w(WL + 9 * 128, BL + 9 * 16, p.dm.vw, p.dm.vb, 8, t, 32);
  __syncthreads();

  // ---- enc2[0] ----
  proj_tiles_wmma(T0, 8, 2, WL + 0 * 128, BL + 0 * 16, T1, t, 0, 2);
  proj_tiles_wmma(T0, 8, 2, WL + 1 * 128, BL + 1 * 16, Kb, t, 0, 2);
  proj_tiles_wmma(T0, 8, 2, WL + 2 * 128, BL + 2 * 16, Vb, t, 0, 2);
  __syncthreads();
  attn_row_inplace(T1, Kb, Vb, 32, t);
  __syncthreads();
  ff_tiles_wmma(T1, WL + 3 * 128, BL + 3 * 16, t, 0, 2);
  __syncthreads();

  // ---- enc2[1] ----
  proj_tiles_wmma(T1, 8, 2, WL + 4 * 128, BL + 4 * 16, T0, t, 0, 2);
  proj_tiles_wmma(T1, 8, 2, WL + 5 * 128, BL + 5 * 16, Kb, t, 0, 2);
  proj_tiles_wmma(T1, 8, 2, WL + 6 * 128, BL + 6 * 16, Vb, t, 0, 2);
  __syncthreads();
  attn_row_inplace(T0, Kb, Vb, 32, t);
  __syncthreads();
  ff_tiles_wmma(T0, WL + 7 * 128, BL + 7 * 16, t, 0, 2);
  __syncthreads();

  // ---- dec2 PMA ----
  proj_tiles_wmma(T0, 8, 2, WL + 8 * 128, BL + 8 * 16, Kb, t, 0, 2);
  proj_tiles_wmma(T0, 8, 2, WL + 9 * 128, BL + 9 * 16, Vb, t, 0, 2);
  if (t < 8) {
    float acc = p.dm.qb[t];
    for (int k = 0; k < 8; ++k) acc += p.S[k] * p.dm.qw[t * 8 + k];
    small[t] = acc;
  }
  __syncthreads();
#pragma unroll
  for (int h = 0; h < NHEADS; ++h)
    sc[h * 32 + t] =
        (small[2 * h] * Kb[t * 8 + 2 * h] + small[2 * h + 1] * Kb[t * 8 + 2 * h + 1]) * ATT_SCALE;
  __syncthreads();
  if (t < NHEADS) {
    const int h = t;
    float m = -1e30f;
    for (int j = 0; j < 32; ++j) m = fmaxf(m, sc[h * 32 + j]);
    float ssum = 0.0f, a0 = 0.0f, a1 = 0.0f;
    for (int j = 0; j < 32; ++j) {
      float e = __expf(sc[h * 32 + j] - m);
      ssum += e;
      a0 += e * Vb[j * 8 + 2 * h];
      a1 += e * Vb[j * 8 + 2 * h + 1];
    }
    const float inv = 1.0f / ssum;
    small[8 + 2 * h]     = small[2 * h]     + a0 * inv;
    small[8 + 2 * h + 1] = small[2 * h + 1] + a1 * inv;
  }
  __syncthreads();
  if (t < 8) {
    float acc = p.dm.ob[t];
    for (int k = 0; k < 8; ++k) acc += small[8 + k] * p.dm.ow[t * 8 + k];
    small[16 + t] = small[8 + t] + fmaxf(acc, 0.0f);
  }
  __syncthreads();
  if (t < 8) {
    float acc = p.lb[t];
    for (int k = 0; k < 8; ++k) acc += small[16 + k] * p.lw[t * 8 + k];
    p.out[b * 264 + 256 + t] = acc;  // z2 tail
  }
}

// ---------------------------------------------------------------------------
extern "C" void kernel_launch(void* const* d_in, const int* in_sizes, int n_in,
                              void* d_out, int out_size, void* d_ws,
                              size_t ws_size, hipStream_t stream) {
  auto F = [&](int i) { return (const float*)d_in[i]; };
  ArgsA a{};
  ArgsB bb{};
  float* zmid = (float*)d_ws;  // 512 x 8 floats

  if (n_in >= 55 && in_sizes[0] > 1000) {
    // Insertion ("dict") order: x, enc1[0..1]{q,k,v,o}{w,b}, dec1{S,mab,lin},
    // enc2[0..1], dec2.
    a.x  = F(0);
    a.e0 = {F(1), F(2), F(3), F(4), F(5), F(6), F(7), F(8)};
    a.e1 = {F(9), F(10), F(11), F(12), F(13), F(14), F(15), F(16)};
    a.S  = F(17);
    a.dm = {F(18), F(19), F(20), F(21), F(22), F(23), F(24), F(25)};
    a.lw = F(26); a.lb = F(27);
    bb.e0 = {F(28), F(29), F(30), F(31), F(32), F(33), F(34), F(35)};
    bb.e1 = {F(36), F(37), F(38), F(39), F(40), F(41), F(42), F(43)};
    bb.S  = F(44);
    bb.dm = {F(45), F(46), F(47), F(48), F(49), F(50), F(51), F(52)};
    bb.lw = F(53); bb.lb = F(54);
  } else {
    // jax pytree sorted-key order: dec1, dec2, enc1, enc2, x (b before w;
    // k,o,q,v).
    a.S  = F(0); a.lb = F(1); a.lw = F(2);
    a.dm = {F(8), F(7), F(4), F(3), F(10), F(9), F(6), F(5)};
    bb.S = F(11); bb.lb = F(12); bb.lw = F(13);
    bb.dm = {F(19), F(18), F(15), F(14), F(21), F(20), F(17), F(16)};
    a.e0 = {F(27), F(26), F(23), F(22), F(29), F(28), F(25), F(24)};
    a.e1 = {F(35), F(34), F(31), F(30), F(37), F(36), F(33), F(32)};
    bb.e0 = {F(43), F(42), F(39), F(38), F(45), F(44), F(41), F(40)};
    bb.e1 = {F(51), F(50), F(47), F(46), F(53), F(52), F(49), F(48)};
    a.x = F(54);
  }
  a.out  = (float*)d_out;
  a.zmid = zmid;
  bb.zmid = zmid;
  bb.out  = (float*)d_out;

  phaseA<<<512, 256, 0, stream>>>(a);
  phaseB<<<16, 32, 0, stream>>>(bb);
}